// DenseCapsule_28741921145148
// MI455X (gfx1250) — compile-verified
//
#include <hip/hip_runtime.h>
#include <math.h>

typedef __attribute__((ext_vector_type(2))) float v2f;
typedef __attribute__((ext_vector_type(8))) float v8f;

namespace {
constexpr int kB = 128, kI = 2048, kE = 8, kO = 16, kD = 16;
constexpr int kBT  = 16;          // batch tile = WMMA M
constexpr int kIC  = 64;          // in-capsules per block
constexpr int kNBT = kB / kBT;    // 8
constexpr int kNIC = kI / kIC;    // 32
constexpr int kSOD = kB * kO * kD;  // 32768 floats (s / out_r size)
}

// XOR-shuffle via ds_swizzle with an immediate pattern: no address VGPRs,
// no bounds-check VALU (group-of-32 mode: and=0x1F, or=0, xor=MASK).
template <int MASK>
__device__ __forceinline__ float swz_xor(float v) {
  constexpr int pat = (MASK << 10) | 0x1F;
  return __int_as_float(__builtin_amdgcn_ds_swizzle(__float_as_int(v), pat));
}

__global__ void caps_zero(float* __restrict__ s) {
  const int idx = blockIdx.x * blockDim.x + threadIdx.x;
  if (idx < kSOD) s[idx] = 0.0f;
}

// One routing pass: recompute x_hat tiles via V_WMMA_F32_16X16X4_F32,
// rebuild logits from prior outputs, softmax over O in LDS, accumulate
// s[b,o,d] = sum_i c * x_hat via global float atomics.
__global__ __launch_bounds__(256, 1) void caps_accum(
    const float* __restrict__ x,     // [B, I, E]
    const float* __restrict__ w,     // [O, I, D, E]
    const float* __restrict__ out0,  // [B, O, D] (valid if nprev>=1)
    const float* __restrict__ out1,  // [B, O, D] (valid if nprev>=2)
    float* __restrict__ s,           // [B, O, D] accumulator
    int nprev)
{
  const int bt   = blockIdx.x;      // batch tile
  const int ic   = blockIdx.y;      // I chunk
  const int tid  = threadIdx.x;
  const int wave = tid >> 5;        // 0..7 -> owns o = 2*wave, 2*wave+1
  const int lane = tid & 31;
  const int lg   = lane & 15;       // lane-in-group (N / M index)
  const int hi   = lane >> 4;       // half-wave select
  const int o0   = wave * 2;

  __shared__ float bmat[kO][kBT];   // routing logits  [o][m]
  __shared__ float cmat[kO][kBT];   // softmax coeffs  [o][m]

  // Preload sum of previous round outputs in the WMMA C/D layout:
  // VGPR v, lane -> (m = v + 8*hi, d = lg). b-logits are linear in out_r,
  // so out0+out1 suffices.
  float opsum[2][8];
  if (nprev > 0) {
#pragma unroll
    for (int ol = 0; ol < 2; ++ol) {
      const int o = o0 + ol;
#pragma unroll
      for (int v = 0; v < 8; ++v) {
        const int gb  = bt * kBT + v + hi * 8;
        const int idx = (gb * kO + o) * kD + lg;
        float t = out0[idx];
        if (nprev > 1) t += out1[idx];
        opsum[ol][v] = t;
      }
    }
  }

  float acc[2][8];
#pragma unroll
  for (int ol = 0; ol < 2; ++ol)
#pragma unroll
    for (int v = 0; v < 8; ++v) acc[ol][v] = 0.0f;

  const bool b3 = (lg & 8) != 0;
  const bool b2 = (lg & 4) != 0;
  const bool b1 = (lg & 2) != 0;

  for (int il = 0; il < kIC; ++il) {
    const int i = ic * kIC + il;

    // A operand (x tile, shared by both o's of this wave).
    // 32-bit 16x4 A layout: lane -> M = lg; half selects K-pair (e).
    const float* xp = x + ((size_t)(bt * kBT + lg) * kI + i) * kE + hi * 2;
    v2f a0 = { xp[0], xp[1] };   // e = 2*hi, 2*hi+1
    v2f a1 = { xp[4], xp[5] };   // e = 4+2*hi, 4+2*hi+1

    float xhf[2][8];
#pragma unroll
    for (int ol = 0; ol < 2; ++ol) {
      const int o = o0 + ol;
      // 32-bit 4x16 B layout: lane -> N = lg (= d); half selects K-pair (e).
      const float* wp = w + (((size_t)(o * kI + i) * kD + lg) * kE) + hi * 2;
      v2f b0v = { wp[0], wp[1] };
      v2f b1v = { wp[4], wp[5] };
      v8f c = {0.f, 0.f, 0.f, 0.f, 0.f, 0.f, 0.f, 0.f};
      // D = A(16x4) x B(4x16) + C ; two K-steps cover E=8
      c = __builtin_amdgcn_wmma_f32_16x16x4_f32(false, a0, false, b0v,
                                                (short)0, c, false, false);
      c = __builtin_amdgcn_wmma_f32_16x16x4_f32(false, a1, false, b1v,
                                                (short)0, c, false, false);
#pragma unroll
      for (int v = 0; v < 8; ++v) xhf[ol][v] = c[v];
    }

    if (nprev == 0) {
      // b == 0 -> c = 1/O uniformly; skip LDS entirely.
      constexpr float cv = 1.0f / kO;
#pragma unroll
      for (int ol = 0; ol < 2; ++ol)
#pragma unroll
        for (int v = 0; v < 8; ++v) acc[ol][v] += cv * xhf[ol][v];
    } else {
      // Logits: b[m,o,i] = sum_d opsum[m,o,d] * x_hat[m,o,i,d].
      // Multi-reduction tree: 8 regs x 16 lanes -> 8 sums in 8 swizzles.
      // After the tree, lane lg holds the sum for v = lg>>1 (m = v + 8*hi).
#pragma unroll
      for (int ol = 0; ol < 2; ++ol) {
        float p[8];
#pragma unroll
        for (int v = 0; v < 8; ++v) p[v] = xhf[ol][v] * opsum[ol][v];
        // step 1 (mask 8): fold v and v+4, role chosen by lane bit3
        float q[4];
#pragma unroll
        for (int j = 0; j < 4; ++j) {
          const float sel = b3 ? p[j + 4] : p[j];
          const float oth = b3 ? p[j] : p[j + 4];
          q[j] = sel + swz_xor<8>(oth);
        }
        // step 2 (mask 4): fold j and j+2, role by bit2
        float r2[2];
#pragma unroll
        for (int j = 0; j < 2; ++j) {
          const float sel = b2 ? q[j + 2] : q[j];
          const float oth = b2 ? q[j] : q[j + 2];
          r2[j] = sel + swz_xor<4>(oth);
        }
        // step 3 (mask 2): fold the last pair, role by bit1
        {
          const float sel = b1 ? r2[1] : r2[0];
          const float oth = b1 ? r2[0] : r2[1];
          float t = sel + swz_xor<2>(oth);
          // step 4 (mask 1): complete the 16-lane sum
          t += swz_xor<1>(t);
          if ((lg & 1) == 0) bmat[o0 + ol][(lg >> 1) + hi * 8] = t;
        }
      }
      __syncthreads();
      {
        // Softmax over the 16 o's per (m, i); thread (o_s, m_s) owns one cell.
        const int o_s = tid >> 4;
        const int m_s = tid & 15;
        float bv[kO];
        float mx = -3.0e38f;
#pragma unroll
        for (int oo = 0; oo < kO; ++oo) {
          bv[oo] = bmat[oo][m_s];
          mx = fmaxf(mx, bv[oo]);
        }
        float den = 0.0f, mine = 0.0f;
#pragma unroll
        for (int oo = 0; oo < kO; ++oo) {
          const float e = __expf(bv[oo] - mx);
          den += e;
          if (oo == o_s) mine = e;
        }
        cmat[o_s][m_s] = mine / den;
      }
      __syncthreads();
#pragma unroll
      for (int ol = 0; ol < 2; ++ol)
#pragma unroll
        for (int v = 0; v < 8; ++v)
          acc[ol][v] += cmat[o0 + ol][v + hi * 8] * xhf[ol][v];
    }
  }

  // Cross-I-chunk reduction (32 writers per address over the pass).
#pragma unroll
  for (int ol = 0; ol < 2; ++ol) {
    const int o = o0 + ol;
#pragma unroll
    for (int v = 0; v < 8; ++v) {
      const int gb = bt * kBT + v + hi * 8;
      atomicAdd(&s[(gb * kO + o) * kD + lg], acc[ol][v]);
    }
  }
}

// squash(s) -> dst ; 16 consecutive d's per (b,o) align with 16-lane groups.
__global__ void caps_squash(const float* __restrict__ s, float* __restrict__ dst) {
  const int idx = blockIdx.x * 256 + threadIdx.x;
  const float v = s[idx];
  float n2 = v * v;
  n2 += swz_xor<1>(n2);
  n2 += swz_xor<2>(n2);
  n2 += swz_xor<4>(n2);
  n2 += swz_xor<8>(n2);
  const float n = sqrtf(n2);
  const float scale = n2 / ((1.0f + n2) * (n + 1e-8f));
  dst[idx] = scale * v;
}

extern "C" void kernel_launch(void* const* d_in, const int* in_sizes, int n_in,
                              void* d_out, int out_size, void* d_ws, size_t ws_size,
                              hipStream_t stream) {
  const float* x = (const float*)d_in[0];   // [128, 2048, 8]
  const float* w = (const float*)d_in[1];   // [16, 2048, 16, 8]
  float* s    = (float*)d_ws;               // [B,O,D] accumulator
  float* out0 = s + kSOD;                   // round-0 outputs
  float* out1 = out0 + kSOD;                // round-1 outputs
  // workspace use: 3 * 32768 * 4B = 384 KB

  const dim3 grid(kNBT, kNIC);
  for (int r = 0; r < 3; ++r) {
    caps_zero<<<(kSOD + 255) / 256, 256, 0, stream>>>(s);
    caps_accum<<<grid, 256, 0, stream>>>(x, w, out0, out1, s, r);
    float* dst = (r == 0) ? out0 : (r == 1) ? out1 : (float*)d_out;
    caps_squash<<<kSOD / 256, 256, 0, stream>>>(s, dst);
  }
}